// CompGCNConv_34394098106413
// MI455X (gfx1250) — compile-verified
//
#include <hip/hip_runtime.h>
#include <hip/hip_bf16.h>

typedef __attribute__((ext_vector_type(2))) float v2f;
typedef __attribute__((ext_vector_type(8))) float v8f;

#define KPI 2.0943951023931953f  /* PI / 1.5 : r = rel / (1.5/PI) */

// ---------------------------------------------------------------------------
// W' = T @ w_loop, where t = rel_transform(x, loop_rel) == x @ T.
//   row k    (k<64): W'[k]    = cos(r_k)*W[k] + sin(r_k)*W[64+k]
//   row 64+k       : W'[64+k] = sin(r_k)*W[k] - cos(r_k)*W[64+k]
// 128x128 elements, one thread each.
// ---------------------------------------------------------------------------
__global__ void build_wprime_kernel(const float* __restrict__ loop_rel,
                                    const float* __restrict__ w_loop,
                                    float* __restrict__ wp) {
    int idx = blockIdx.x * blockDim.x + threadIdx.x;
    if (idx >= 128 * 128) return;
    int r = idx >> 7;      // output row
    int n = idx & 127;     // output col
    int k = r & 63;
    float rr = loop_rel[k] * KPI;
    float cr = cosf(rr), sr = sinf(rr);
    float w0 = w_loop[k * 128 + n];
    float w1 = w_loop[(64 + k) * 128 + n];
    wp[idx] = (r < 64) ? (cr * w0 + sr * w1) : (sr * w0 - cr * w1);
}

// ---------------------------------------------------------------------------
// deg[row] += 1 for each edge (in-half -> deg_in, out-half -> deg_out)
// edge_index flat layout: [0,E2) = row indices, [E2,2*E2) = col indices
// ---------------------------------------------------------------------------
__global__ void degree_kernel(const int* __restrict__ edge_index,
                              float* __restrict__ deg_in,
                              float* __restrict__ deg_out,
                              int E, int E2) {
    int e = blockIdx.x * blockDim.x + threadIdx.x;
    if (e >= E2) return;
    int row = edge_index[e];
    float* deg = (e < E) ? deg_in : deg_out;
    atomicAdd(deg + row, 1.0f);
}

// deg -> deg^-0.5 (0 where deg==0); deg_in/deg_out are contiguous
__global__ void dinv_kernel(float* __restrict__ deg, int n) {
    int i = blockIdx.x * blockDim.x + threadIdx.x;
    if (i < n) {
        float d = deg[i];
        deg[i] = (d > 0.0f) ? (1.0f / sqrtf(d)) : 0.0f;
    }
}

// ---------------------------------------------------------------------------
// Per-edge: acc[row] += norm * rel_transform(x[col], rel_embed[et])
// One wave32 per edge; each lane handles 2 (re,im) pairs -> 4 output floats.
// ---------------------------------------------------------------------------
__global__ void scatter_kernel(const float* __restrict__ x,
                               const int* __restrict__ edge_index,
                               const int* __restrict__ edge_type,
                               const float* __restrict__ rel_embed,
                               const float* __restrict__ dinv_in,
                               const float* __restrict__ dinv_out,
                               float* __restrict__ acc_in,
                               float* __restrict__ acc_out,
                               int E, int E2) {
    int gid  = blockIdx.x * blockDim.x + threadIdx.x;
    int e    = gid >> 5;          // one wave per edge
    int lane = threadIdx.x & 31;
    if (e >= E2) return;

    int row = edge_index[e];
    int col = edge_index[E2 + e];
    int et  = edge_type[e];
    const float* dinv = (e < E) ? dinv_in : dinv_out;
    float* acc        = (e < E) ? acc_in  : acc_out;
    float norm = dinv[row] * dinv[col];

    int k = lane * 2;                         // k in [0,64), 2 pairs per lane
    const float* relp = rel_embed + (size_t)et * 64 + k;
    float r0 = relp[0] * KPI;
    float r1 = relp[1] * KPI;
    float c0 = cosf(r0), s0 = sinf(r0);
    float c1 = cosf(r1), s1 = sinf(r1);

    const float* xp = x + (size_t)col * 128;
    float re0 = xp[k],      re1 = xp[k + 1];
    float im0 = xp[64 + k], im1 = xp[64 + k + 1];

    float o0 = (re0 * c0 + im0 * s0) * norm;  // column k
    float o1 = (re1 * c1 + im1 * s1) * norm;  // column k+1
    float o2 = (re0 * s0 - im0 * c0) * norm;  // column 64+k
    float o3 = (re1 * s1 - im1 * c1) * norm;  // column 64+k+1

    float* ap = acc + (size_t)row * 128;
    atomicAdd(ap + k,          o0);
    atomicAdd(ap + k + 1,      o1);
    atomicAdd(ap + 64 + k,     o2);
    atomicAdd(ap + 64 + k + 1, o3);
}

// ---------------------------------------------------------------------------
// Accumulate a 16x32 tile of A[M,128] @ W[128,128] into (c0,c1) via
// V_WMMA_F32_16X16X4_F32. Branch-free: EXEC stays all-1s.
// A fragment (ISA 7.12.2, 32-bit A 16x4): lane 0-15 M=lr, vgprs hold
// K = {k0,k0+1} (lo half) / {k0+2,k0+3} (hi half). B mirrored on N.
// ---------------------------------------------------------------------------
__device__ __forceinline__ void gemm_pair(v8f& c0, v8f& c1,
                                          const float* __restrict__ A,
                                          const float* __restrict__ W,
                                          int m0, int n0, int lr, int hi) {
    const float* Arow = A + (size_t)(m0 + lr) * 128;
#pragma unroll 4
    for (int k0 = 0; k0 < 128; k0 += 4) {
        int ka = k0 + 2 * hi;
        v2f a = *(const v2f*)(Arow + ka);          // 8B load, ka is even
        const float* Wr0 = W + (size_t)ka * 128 + lr;
        const float* Wr1 = Wr0 + 128;
        v2f b0, b1;
        b0.x = Wr0[n0];      b0.y = Wr1[n0];
        b1.x = Wr0[n0 + 16]; b1.y = Wr1[n0 + 16];
        c0 = __builtin_amdgcn_wmma_f32_16x16x4_f32(
                 false, a, false, b0, (short)0, c0, false, false);
        c1 = __builtin_amdgcn_wmma_f32_16x16x4_f32(
                 false, a, false, b1, (short)0, c1, false, false);
    }
}

// ---------------------------------------------------------------------------
// out = (acc_in@w_in + acc_out@w_out + x@W') * 1/3  -- single fused pass.
// One wave per 16x32 tile; M must be a multiple of 16 (100000 = 6250*16).
// ---------------------------------------------------------------------------
__global__ void fused3_gemm_kernel(const float* __restrict__ acc_in,
                                   const float* __restrict__ w_in,
                                   const float* __restrict__ acc_out,
                                   const float* __restrict__ w_out,
                                   const float* __restrict__ x,
                                   const float* __restrict__ wprime,
                                   float* __restrict__ C) {
    int m0   = blockIdx.x * 16;
    int n0   = blockIdx.y * 32;
    int lane = threadIdx.x & 31;
    int lr   = lane & 15;
    int hi   = lane >> 4;

    v8f c0 = {}, c1 = {};
    gemm_pair(c0, c1, acc_in,  w_in,   m0, n0, lr, hi);
    gemm_pair(c0, c1, acc_out, w_out,  m0, n0, lr, hi);
    gemm_pair(c0, c1, x,       wprime, m0, n0, lr, hi);

    const float s = 1.0f / 3.0f;
#pragma unroll
    for (int v = 0; v < 8; ++v) {
        int m = m0 + v + hi * 8;   // C/D layout: vgpr v -> M=v (lo) / v+8 (hi)
        float* p = C + (size_t)m * 128 + lr;
        p[n0]      = c0[v] * s;
        p[n0 + 16] = c1[v] * s;
    }
}

// ---------------------------------------------------------------------------
// C[M,N] = A[M,K] @ W[K,N] with branch-free M-tail guard (for M=500 rel GEMM):
// out-of-range rows read a clamped row and are zeroed by a 0/1 mask.
// ---------------------------------------------------------------------------
__global__ void wmma_gemm_guard_kernel(const float* __restrict__ A,
                                       const float* __restrict__ W,
                                       float* __restrict__ C,
                                       int M, int N, int K) {
    int m0   = blockIdx.x * 16;
    int n0   = blockIdx.y * 16;
    int lane = threadIdx.x & 31;
    int lr   = lane & 15;
    int hi   = lane >> 4;

    int   row = m0 + lr;
    int   rc  = (row < M) ? row : (M - 1);
    float msk = (row < M) ? 1.0f : 0.0f;
    const float* Arow = A + (size_t)rc * K;

    v8f c = {};
#pragma unroll 4
    for (int k0 = 0; k0 < K; k0 += 4) {
        int ka = k0 + 2 * hi;
        v2f a = *(const v2f*)(Arow + ka);
        a.x *= msk;
        a.y *= msk;
        v2f b;
        b.x = W[(size_t)ka * N + n0 + lr];
        b.y = W[(size_t)(ka + 1) * N + n0 + lr];
        c = __builtin_amdgcn_wmma_f32_16x16x4_f32(
                false, a, false, b, (short)0, c, false, false);
    }
#pragma unroll
    for (int v = 0; v < 8; ++v) {
        int m = m0 + v + hi * 8;
        if (m < M) C[(size_t)m * N + n0 + lr] = c[v];
    }
}

// ---------------------------------------------------------------------------
extern "C" void kernel_launch(void* const* d_in, const int* in_sizes, int n_in,
                              void* d_out, int out_size, void* d_ws, size_t ws_size,
                              hipStream_t stream) {
    const float* x          = (const float*)d_in[0];
    const int*   edge_index = (const int*)d_in[1];
    const int*   edge_type  = (const int*)d_in[2];
    const float* rel_embed  = (const float*)d_in[3];
    const float* w_loop     = (const float*)d_in[4];
    const float* w_in       = (const float*)d_in[5];
    const float* w_out      = (const float*)d_in[6];
    const float* w_rel      = (const float*)d_in[7];
    const float* loop_rel   = (const float*)d_in[8];

    const int num_ent = in_sizes[0] / 128;   // 100000 (multiple of 16)
    const int E2      = in_sizes[2];         // 1,000,000
    const int E       = E2 / 2;              // 500,000
    const int num_rel = in_sizes[3] / 64;    // 500

    // workspace: acc_in | acc_out | deg_in | deg_out | wprime  (~103 MB)
    float* acc_in  = (float*)d_ws;
    float* acc_out = acc_in  + (size_t)num_ent * 128;
    float* deg_in  = acc_out + (size_t)num_ent * 128;
    float* deg_out = deg_in  + num_ent;
    float* wprime  = deg_out + num_ent;

    size_t zero_bytes = ((size_t)num_ent * 128 * 2 + (size_t)num_ent * 2) * sizeof(float);
    hipMemsetAsync(d_ws, 0, zero_bytes, stream);

    build_wprime_kernel<<<(128 * 128 + 255) / 256, 256, 0, stream>>>(loop_rel, w_loop, wprime);

    degree_kernel<<<(E2 + 255) / 256, 256, 0, stream>>>(edge_index, deg_in, deg_out, E, E2);
    dinv_kernel<<<(2 * num_ent + 255) / 256, 256, 0, stream>>>(deg_in, 2 * num_ent);

    {   // one wave32 per edge
        long long threads = (long long)E2 * 32;
        int blocks = (int)((threads + 255) / 256);
        scatter_kernel<<<blocks, 256, 0, stream>>>(x, edge_index, edge_type, rel_embed,
                                                   deg_in, deg_out, acc_in, acc_out, E, E2);
    }

    float* out_main = (float*)d_out;                       // [num_ent,128]
    float* out_rel  = out_main + (size_t)num_ent * 128;    // [num_rel,64]

    dim3 gmain(num_ent / 16, 128 / 32);                    // 6250 x 4, 16x32 tiles
    fused3_gemm_kernel<<<gmain, 32, 0, stream>>>(acc_in, w_in, acc_out, w_out,
                                                 x, wprime, out_main);

    dim3 grel((num_rel + 15) / 16, 64 / 16);               // 32 x 4
    wmma_gemm_guard_kernel<<<grel, 32, 0, stream>>>(rel_embed, w_rel, out_rel,
                                                    num_rel, 64, 64);
}